// MultiHeadGraphAttentionLayer_69312182223584
// MI455X (gfx1250) — compile-verified
//
#include <hip/hip_runtime.h>

// out[b, n, h*64+d] = sum_f h[b,n,f] * W[h,f,d]
// (softmax row-sums are exactly 1, so the reference's attention path is identity)
//
// Single GEMM: C[16384, 256] = A[16384, 256] x Wc[256, 256]
//   Wc[f, h*64+d] = W[h, f, d]   (permutation applied while staging W into LDS)

typedef __attribute__((ext_vector_type(2))) float v2f;
typedef __attribute__((ext_vector_type(4))) float v4f;
typedef __attribute__((ext_vector_type(8))) float v8f;

constexpr int BATCH   = 8;
constexpr int N_NODES = 2048;
constexpr int F_IN    = 256;   // K
constexpr int HEADS   = 4;
constexpr int DH      = 64;
constexpr int M_TOTAL = BATCH * N_NODES;   // 16384
constexpr int N_OUT   = HEADS * DH;        // 256

constexpr int BM       = 128;  // C rows per block
constexpr int KC       = 32;   // K chunk staged in LDS
constexpr int STRIDE_A = 36;   // floats; 144B rows: 16B-aligned stores, conflict-free reads
constexpr int STRIDE_BT= 36;   // floats; B stored transposed [n][k], same padding logic

__global__ __launch_bounds__(256) void gat_proj_wmma(
    const float* __restrict__ hmat,
    const float* __restrict__ Wmat,
    float* __restrict__ out)
{
    __shared__ float ldsA [BM    * STRIDE_A ];  // 128*36*4  = 18432 B
    __shared__ float ldsBT[N_OUT * STRIDE_BT];  // 256*36*4  = 36864 B

    const int tid     = threadIdx.x;
    const int lane    = tid & 31;
    const int wave    = tid >> 5;
    const int waveM   = wave & 3;           // 4 waves along M (32 rows each)
    const int waveN   = wave >> 2;          // 2 waves along N (128 cols each)
    const int mBlock  = blockIdx.x * BM;
    const int mBase   = waveM * 32;         // within-block row base
    const int nBase   = waveN * 128;        // column base
    const int halfSel = lane >> 4;          // 0: lanes 0-15, 1: lanes 16-31
    const int l15     = lane & 15;

    const v8f vzero = {0.f, 0.f, 0.f, 0.f, 0.f, 0.f, 0.f, 0.f};
    v8f acc[16];
#pragma unroll
    for (int i = 0; i < 16; ++i) acc[i] = vzero;

    for (int kc = 0; kc < F_IN; kc += KC) {
        __syncthreads();  // protect LDS from previous chunk's readers

        // ---- Stage A chunk [128 x 32] -> LDS (coalesced b128 loads/stores) ----
#pragma unroll
        for (int it = 0; it < 4; ++it) {
            int lin4 = it * 256 + tid;          // 1024 float4 total
            int row  = lin4 >> 3;               // 8 float4 per 32-float row
            int col  = (lin4 & 7) << 2;
            const v4f* g = reinterpret_cast<const v4f*>(
                hmat + (size_t)(mBlock + row) * F_IN + kc + col);
            *reinterpret_cast<v4f*>(&ldsA[row * STRIDE_A + col]) = *g;
        }

        // ---- Stage B chunk TRANSPOSED: ldsBT[n][k] = Wc[kc+k][n] = W[n>>6][kc+k][n&63]
        // Each thread gathers 4 K-adjacent W elements (each slot strided across
        // lanes: consecutive tid -> next k-quad, 8 tids per d-column) and writes one b128 row run.
#pragma unroll
        for (int it = 0; it < 8; ++it) {
            int lin  = it * 256 + tid;          // 2048 (n, kk4) pairs
            int n    = lin >> 3;                // 8 k-quads per n
            int kk4  = (lin & 7) << 2;
            int head = n >> 6;
            int dcol = n & 63;
            const float* g =
                Wmat + (size_t)head * (F_IN * DH) + (size_t)(kc + kk4) * DH + dcol;
            v4f w;
            w.x = g[0 * DH];
            w.y = g[1 * DH];
            w.z = g[2 * DH];
            w.w = g[3 * DH];
            *reinterpret_cast<v4f*>(&ldsBT[n * STRIDE_BT + kk4]) = w;
        }
        __syncthreads();

        // ---- Compute: 8 k-steps x (2 M-tiles x 8 N-tiles) WMMAs ----
#pragma unroll
        for (int kk = 0; kk < KC; kk += 4) {
            // A fragment: 16x4 f32 -> 2 VGPRs.
            // lane<16: K = kk+{0,1}; lane>=16: K = kk+{2,3}; M = lane&15.
            v2f afrag[2];
#pragma unroll
            for (int mt = 0; mt < 2; ++mt) {
                const float* p =
                    &ldsA[(mBase + mt * 16 + l15) * STRIDE_A + kk + 2 * halfSel];
                afrag[mt] = *reinterpret_cast<const v2f*>(p);  // aligned b64
            }
#pragma unroll
            for (int nt = 0; nt < 8; ++nt) {
                // B fragment: 4x16 f32 -> 2 VGPRs; K-adjacent pair is contiguous
                // in the transposed LDS layout -> single aligned ds_load_b64.
                const float* p =
                    &ldsBT[(nBase + nt * 16 + l15) * STRIDE_BT + kk + 2 * halfSel];
                v2f bfrag = *reinterpret_cast<const v2f*>(p);
#pragma unroll
                for (int mt = 0; mt < 2; ++mt) {
                    acc[mt * 8 + nt] = __builtin_amdgcn_wmma_f32_16x16x4_f32(
                        /*neg_a=*/false, afrag[mt],
                        /*neg_b=*/false, bfrag,
                        /*c_mod=*/(short)0, acc[mt * 8 + nt],
                        /*reuse_a=*/false, /*reuse_b=*/false);
                }
            }
        }
    }

    // ---- Store C: VGPR j of each tile holds row M=j (lanes 0-15) / M=j+8 (lanes 16-31) ----
#pragma unroll
    for (int mt = 0; mt < 2; ++mt) {
#pragma unroll
        for (int nt = 0; nt < 8; ++nt) {
            int col = nBase + nt * 16 + l15;
#pragma unroll
            for (int j = 0; j < 8; ++j) {
                int row = mBlock + mBase + mt * 16 + 8 * halfSel + j;
                out[(size_t)row * N_OUT + col] = acc[mt * 8 + nt][j];
            }
        }
    }
}

extern "C" void kernel_launch(void* const* d_in, const int* in_sizes, int n_in,
                              void* d_out, int out_size, void* d_ws, size_t ws_size,
                              hipStream_t stream) {
    (void)in_sizes; (void)n_in; (void)out_size; (void)d_ws; (void)ws_size;
    const float* hmat = (const float*)d_in[0];   // [8, 2048, 256]
    // d_in[1] = adj, unused by the reference math
    const float* Wmat = (const float*)d_in[2];   // [4, 256, 64]
    float* out = (float*)d_out;                  // [8, 2048, 256]

    dim3 grid(M_TOTAL / BM);   // 128 blocks
    dim3 block(256);           // 8 waves (wave32)
    gat_proj_wmma<<<grid, block, 0, stream>>>(hmat, Wmat, out);
}